// PerNodeValidMLP_6588479832304
// MI455X (gfx1250) — compile-verified
//
#include <hip/hip_runtime.h>

typedef __attribute__((ext_vector_type(2))) float v2f;
typedef __attribute__((ext_vector_type(8))) float v8f;

#define N_SAMPLES 262144
#define N_NODES   24
#define IN_CH     32
#define HID       32
#define TILE_M    16
#define TILES_PER_WAVE  16
#define WAVES_PER_BLOCK 8

// Waves per node = (262144/16)/16 = 1024; total waves = 24*1024 = 24576
#define WAVES_PER_NODE  ((N_SAMPLES / TILE_M) / TILES_PER_WAVE)
#define TOTAL_BLOCKS    ((N_NODES * WAVES_PER_NODE) / WAVES_PER_BLOCK)

__global__ __launch_bounds__(256) void PerNodeValidMLP_kernel(
    const float* __restrict__ h,     // [S, N, 32]
    const int*   __restrict__ valid, // [S, N]
    const float* __restrict__ W1,    // [N, 32, 32]
    const float* __restrict__ b1,    // [N, 32]
    const float* __restrict__ W2,    // [N, 32, 1]
    const float* __restrict__ b2,    // [N]
    float*       __restrict__ out)   // [S, N, 1]
{
    const int lane = threadIdx.x & 31;
    const int wave = blockIdx.x * WAVES_PER_BLOCK + (threadIdx.x >> 5);
    const int node = wave / WAVES_PER_NODE;
    const int wn   = wave % WAVES_PER_NODE;
    const int col  = lane & 15;   // column within 16-wide tile (B/C/D lane dim)
    const int hi   = lane >> 4;   // half-wave selector (K pair for A/B layouts)

    // ---- Load per-node weights once into registers ----
    // B layout for V_WMMA_F32_16X16X4_F32 (4x16, K x N):
    //   VGPR0 (.x): lanes 0-15 row K=kb+0, lanes 16-31 row K=kb+2
    //   VGPR1 (.y): lanes 0-15 row K=kb+1, lanes 16-31 row K=kb+3
    v2f Bv[2][8];
    const float* w1n = W1 + (size_t)node * (IN_CH * HID);
    #pragma unroll
    for (int t = 0; t < 2; ++t) {
        #pragma unroll
        for (int k8 = 0; k8 < 8; ++k8) {
            const int kb = k8 * 4;
            const int c  = t * 16 + col;
            Bv[t][k8].x = w1n[(kb + 0 + 2 * hi) * HID + c];
            Bv[t][k8].y = w1n[(kb + 1 + 2 * hi) * HID + c];
        }
    }
    const float b1v0 = b1[node * HID + col];        // bias for hidden col (tile 0)
    const float b1v1 = b1[node * HID + 16 + col];   // tile 1
    const float w2v0 = W2[node * HID + col];        // W2 for hidden col (tile 0)
    const float w2v1 = W2[node * HID + 16 + col];   // tile 1
    const float b2s  = b2[node];

    // Output-row mapping for the final predicated store:
    // res[r]: lanes 0-15 hold row r, lanes 16-31 hold row r+8 (D layout).
    // Lane l (with l&8==0) stores row (l&7) + 8*hi.
    const int rowl = (lane & 7) + 8 * hi;

    for (int tIdx = 0; tIdx < TILES_PER_WAVE; ++tIdx) {
        const int base = (wn * TILES_PER_WAVE + tIdx) * TILE_M;
        const int s    = base + col;  // sample index for this lane's A rows
        const float* hrow = h + ((size_t)s * N_NODES + node) * IN_CH;

        // C initialized to b1 broadcast (bias depends only on column = lane)
        v8f acc0, acc1;
        #pragma unroll
        for (int r = 0; r < 8; ++r) { acc0[r] = b1v0; acc1[r] = b1v1; }

        // Layer 1: 8 K-steps of 4, two 16-wide hidden tiles
        #pragma unroll
        for (int k8 = 0; k8 < 8; ++k8) {
            // A layout: lanes 0-15 K=kb,kb+1 ; lanes 16-31 K=kb+2,kb+3
            v2f a = *(const v2f*)(hrow + k8 * 4 + 2 * hi);
            a.x = fmaxf(a.x, 0.0f);   // x = relu(h)
            a.y = fmaxf(a.y, 0.0f);
            acc0 = __builtin_amdgcn_wmma_f32_16x16x4_f32(
                false, a, false, Bv[0][k8], (short)0, acc0, false, false);
            acc1 = __builtin_amdgcn_wmma_f32_16x16x4_f32(
                false, a, false, Bv[1][k8], (short)0, acc1, false, false);
        }

        // Layer 2: out[m] = sum_w relu(hid[m][w]) * W2[w]  (+ b2, masked)
        float res[8];
        #pragma unroll
        for (int r = 0; r < 8; ++r) {
            float v = fmaxf(acc0[r], 0.0f) * w2v0 + fmaxf(acc1[r], 0.0f) * w2v1;
            // butterfly reduce over 16 lanes; masks <16 stay within each half
            v += __shfl_xor(v, 1, 32);
            v += __shfl_xor(v, 2, 32);
            v += __shfl_xor(v, 4, 32);
            v += __shfl_xor(v, 8, 32);
            res[r] = v;
        }

        // Select this lane's row result (r == lane&7) via cndmask chain
        const int rsel = lane & 7;
        float rv = res[0];
        rv = (rsel == 1) ? res[1] : rv;
        rv = (rsel == 2) ? res[2] : rv;
        rv = (rsel == 3) ? res[3] : rv;
        rv = (rsel == 4) ? res[4] : rv;
        rv = (rsel == 5) ? res[5] : rv;
        rv = (rsel == 6) ? res[6] : rv;
        rv = (rsel == 7) ? res[7] : rv;

        const size_t oidx = (size_t)(base + rowl) * N_NODES + node;
        const int vflag = valid[oidx];
        const float ov  = (vflag > 0) ? (rv + b2s) : 0.0f;
        if ((lane & 8) == 0) {
            out[oidx] = ov;
        }
    }
}

extern "C" void kernel_launch(void* const* d_in, const int* in_sizes, int n_in,
                              void* d_out, int out_size, void* d_ws, size_t ws_size,
                              hipStream_t stream) {
    (void)in_sizes; (void)n_in; (void)d_ws; (void)ws_size; (void)out_size;
    const float* h     = (const float*)d_in[0];
    const int*   valid = (const int*)d_in[1];
    const float* W1    = (const float*)d_in[2];
    const float* b1    = (const float*)d_in[3];
    const float* W2    = (const float*)d_in[4];
    const float* b2    = (const float*)d_in[5];
    float* out = (float*)d_out;

    PerNodeValidMLP_kernel<<<TOTAL_BLOCKS, 256, 0, stream>>>(
        h, valid, W1, b1, W2, b2, out);
}